// recon_encoder_59571196395530
// MI455X (gfx1250) — compile-verified
//
#include <hip/hip_runtime.h>
#include <hip/hip_bf16.h>
#include <math.h>

#define N_NODES   50000
#define N_EDGES   800000
#define D_IN      128
#define HID       128
#define LAT       64
#define NEG_SLOPE 0.2f

typedef __attribute__((ext_vector_type(2))) float v2f;
typedef __attribute__((ext_vector_type(8))) float v8f;

// ---------------------------------------------------------------- fill
__global__ void fill_kernel(float* __restrict__ p, float v, int n) {
  int i = blockIdx.x * blockDim.x + threadIdx.x;
  if (i < n) p[i] = v;
}

// ---------------------------------------------------------------- GEMM
// Out[N x KOUT] = A[N x KIN] * W[KIN x KOUT], fp32, V_WMMA_F32_16X16X4_F32.
// Block = 128 threads (4 waves), one 16-row strip per block, LDS-staged A.
// Compile-time KOUT -> all B-load / store offsets are immediates.
// Per wave: KOUT/64 column tiles interleaved in the K loop, each with two
// independent accumulator chains (even/odd K groups) for WMMA ILP.
template <int KIN, int KOUT>
__global__ __launch_bounds__(128) void gemm_wmma_f32(const float* __restrict__ A,
                                                     const float* __restrict__ W,
                                                     float* __restrict__ Out) {
  constexpr int LDSS = KIN + 4;   // padded stride -> conflict-free column reads
  constexpr int NT   = KOUT / 64; // column tiles per wave (1 or 2)
  __shared__ float As[16 * LDSS];
  const int m0 = blockIdx.x * 16;
  const int t  = threadIdx.x;

  // cooperative, coalesced load of the 16 x KIN A strip
  for (int p = t; p < 16 * KIN; p += 128) {
    int r = p / KIN, c = p % KIN;
    As[r * LDSS + c] = A[(size_t)(m0 + r) * KIN + c];
  }
  __syncthreads();

  const int wave = t >> 5;
  const int lane = t & 31;
  const int half = lane >> 4;     // 0 -> K pair {0,1}; 1 -> {2,3}
  const int l16  = lane & 15;

  // per-lane base pointers; everything below uses immediate offsets
  const float* wp = W + (size_t)(2 * half) * KOUT + wave * 16 + l16;
  const float* ap = &As[l16 * LDSS + 2 * half];

  v8f acc0[NT] = {}, acc1[NT] = {};

#pragma unroll
  for (int k0 = 0; k0 < KIN; k0 += 8) {
    v2f a0, a1;
    // A 16x4 layout: lanes 0-15: K=k0+{0,1}; lanes 16-31: K=k0+{2,3}
    a0.x = ap[k0];     a0.y = ap[k0 + 1];
    a1.x = ap[k0 + 4]; a1.y = ap[k0 + 5];
#pragma unroll
    for (int i = 0; i < NT; ++i) {
      v2f b0, b1;
      // B 4x16 layout (K rows striped across lanes within a VGPR)
      b0.x = wp[(k0    ) * KOUT + i * 64];
      b0.y = wp[(k0 + 1) * KOUT + i * 64];
      b1.x = wp[(k0 + 4) * KOUT + i * 64];
      b1.y = wp[(k0 + 5) * KOUT + i * 64];
      acc0[i] = __builtin_amdgcn_wmma_f32_16x16x4_f32(false, a0, false, b0,
                                                      (short)0, acc0[i], false, false);
      acc1[i] = __builtin_amdgcn_wmma_f32_16x16x4_f32(false, a1, false, b1,
                                                      (short)0, acc1[i], false, false);
    }
  }

  // C/D layout: VGPR r -> M = r (lanes 0-15) / r+8 (lanes 16-31), N = lane&15
#pragma unroll
  for (int i = 0; i < NT; ++i) {
    float* op = Out + (size_t)(m0 + half * 8) * KOUT + wave * 16 + i * 64 + l16;
#pragma unroll
    for (int r = 0; r < 8; ++r)
      op[(size_t)r * KOUT] = acc0[i][r] + acc1[i][r];
  }
}

// ---------------------------------------------------------------- row-dot
// out[n] = dot(H[n, :C], a[:C]) ; one wave32 per node
template <int C>
__global__ void rowdot_kernel(const float* __restrict__ H,
                              const float* __restrict__ a,
                              float* __restrict__ out) {
  int warp = (blockIdx.x * blockDim.x + threadIdx.x) >> 5;
  int lane = threadIdx.x & 31;
  if (warp >= N_NODES) return;
  float s = 0.f;
#pragma unroll
  for (int c = lane; c < C; c += 32)
    s += H[(size_t)warp * C + c] * a[c];
#pragma unroll
  for (int off = 16; off; off >>= 1)
    s += __shfl_xor(s, off, 32);
  if (lane == 0) out[warp] = s;
}

// ---------------------------------------------------------------- edge passes
__device__ inline void atomicMaxFloat(float* addr, float v) {
  // sign-split trick -> hardware global_atomic_max_i32 / global_atomic_min_u32
  if (v >= 0.f) atomicMax((int*)addr, __float_as_int(v));
  else          atomicMin((unsigned int*)addr, __float_as_uint(v));
}

__global__ void edge_pass1(const int* __restrict__ src, const int* __restrict__ dst,
                           const float* __restrict__ as_, const float* __restrict__ ad_,
                           float* __restrict__ eout, float* __restrict__ maxb) {
  int e = blockIdx.x * blockDim.x + threadIdx.x;
  if (e >= N_EDGES) return;
  int s = src[e], d = dst[e];
  float v = as_[s] + ad_[d];
  v = v > 0.f ? v : NEG_SLOPE * v;            // leaky_relu
  eout[e] = v;
  atomicMaxFloat(&maxb[d], v);
}

__global__ void edge_pass2(const int* __restrict__ dst, float* __restrict__ e,
                           const float* __restrict__ maxb, float* __restrict__ denom) {
  int i = blockIdx.x * blockDim.x + threadIdx.x;
  if (i >= N_EDGES) return;
  int d = dst[i];
  float ex = expf(e[i] - maxb[d]);
  e[i] = ex;
  atomicAdd(&denom[d], ex);
}

// one wave32 per edge; lanes cover channels; f32 atomic scatter-add (L2 resident)
template <int C>
__global__ void edge_pass3(const int* __restrict__ src, const int* __restrict__ dst,
                           const float* __restrict__ ex, const float* __restrict__ denom,
                           const float* __restrict__ hs, float* __restrict__ aggr) {
  int warp = (blockIdx.x * blockDim.x + threadIdx.x) >> 5;
  int lane = threadIdx.x & 31;
  if (warp >= N_EDGES) return;
  int s = src[warp], d = dst[warp];
  float alpha = ex[warp] / (denom[d] + 1e-16f);
#pragma unroll
  for (int c = lane; c < C; c += 32)
    atomicAdd(&aggr[(size_t)d * C + c], hs[(size_t)s * C + c] * alpha);
}

// ---------------------------------------------------------------- elementwise
__global__ void combine_z(const float* __restrict__ aggr, const float* __restrict__ b1,
                          const float* __restrict__ xl, const float* __restrict__ lb,
                          float* __restrict__ z) {
  int i = blockIdx.x * blockDim.x + threadIdx.x;
  if (i >= N_NODES * HID) return;
  int c = i & (HID - 1);
  float v = aggr[i] + b1[c] + xl[i] + lb[c];
  z[i] = v > 0.f ? v : 0.f;                   // relu(gat + x@lin1_w + lin1_b)
}

__global__ void final_out(const float* __restrict__ aggr, const float* __restrict__ b2,
                          float* __restrict__ out) {
  int i = blockIdx.x * blockDim.x + threadIdx.x;
  if (i >= N_NODES * LAT) return;
  int c = i & (LAT - 1);
  out[i] = aggr[i] + b2[c];
}

// ---------------------------------------------------------------- launch
extern "C" void kernel_launch(void* const* d_in, const int* in_sizes, int n_in,
                              void* d_out, int out_size, void* d_ws, size_t ws_size,
                              hipStream_t stream) {
  (void)in_sizes; (void)n_in; (void)out_size; (void)ws_size;
  const float* x      = (const float*)d_in[0];
  const int*   eidx   = (const int*)d_in[1];          // [2, E] row-major
  const float* lin1_w = (const float*)d_in[2];
  const float* lin1_b = (const float*)d_in[3];
  const float* w1_src = (const float*)d_in[4];
  const float* w1_dst = (const float*)d_in[5];
  const float* a1_src = (const float*)d_in[6];
  const float* a1_dst = (const float*)d_in[7];
  const float* b1     = (const float*)d_in[8];
  const float* w2_src = (const float*)d_in[9];
  const float* w2_dst = (const float*)d_in[10];
  const float* a2_src = (const float*)d_in[11];
  const float* a2_dst = (const float*)d_in[12];
  const float* b2     = (const float*)d_in[13];
  const int* src = eidx;
  const int* dst = eidx + N_EDGES;

  // workspace layout (floats)
  float* w = (float*)d_ws;
  float* hs     = w;                   w += (size_t)N_NODES * HID;  // reused as hs2 [N,64]
  float* hd     = w;                   w += (size_t)N_NODES * HID;  // reused as hd2
  float* xl     = w;                   w += (size_t)N_NODES * HID;
  float* aggr   = w;                   w += (size_t)N_NODES * HID;  // reused as aggr2
  float* z      = w;                   w += (size_t)N_NODES * HID;
  float* alphaS = w;                   w += N_NODES;
  float* alphaD = w;                   w += N_NODES;
  float* maxb   = w;                   w += N_NODES;
  float* denom  = w;                   w += N_NODES;
  float* edgeE  = w;                   w += N_EDGES;

  const int gemmGrid = N_NODES / 16;             // 3125 (exact)
  const int dotGrid  = (N_NODES + 7) / 8;        // 8 waves/block
  const int edgGrid  = (N_EDGES + 255) / 256;
  const int e3Grid   = N_EDGES / 8;              // 1 wave per edge, 8/block (exact)
  const float NEGINF = -__builtin_inff();

  // ---- layer 1: dense transforms (WMMA fp32) ----
  gemm_wmma_f32<D_IN, HID><<<gemmGrid, 128, 0, stream>>>(x, w1_src, hs);
  gemm_wmma_f32<D_IN, HID><<<gemmGrid, 128, 0, stream>>>(x, w1_dst, hd);
  gemm_wmma_f32<D_IN, HID><<<gemmGrid, 128, 0, stream>>>(x, lin1_w, xl);
  rowdot_kernel<HID><<<dotGrid, 256, 0, stream>>>(hs, a1_src, alphaS);
  rowdot_kernel<HID><<<dotGrid, 256, 0, stream>>>(hd, a1_dst, alphaD);

  // ---- layer 1: segment softmax + aggregation ----
  fill_kernel<<<(N_NODES + 255) / 256, 256, 0, stream>>>(maxb, NEGINF, N_NODES);
  fill_kernel<<<(N_NODES + 255) / 256, 256, 0, stream>>>(denom, 0.f, N_NODES);
  fill_kernel<<<((int)(N_NODES * HID) + 255) / 256, 256, 0, stream>>>(aggr, 0.f, N_NODES * HID);
  edge_pass1<<<edgGrid, 256, 0, stream>>>(src, dst, alphaS, alphaD, edgeE, maxb);
  edge_pass2<<<edgGrid, 256, 0, stream>>>(dst, edgeE, maxb, denom);
  edge_pass3<HID><<<e3Grid, 256, 0, stream>>>(src, dst, edgeE, denom, hs, aggr);

  // z = relu(aggr + b1 + x@lin1_w + lin1_b)
  combine_z<<<((int)(N_NODES * HID) + 255) / 256, 256, 0, stream>>>(aggr, b1, xl, lin1_b, z);

  // ---- layer 2 ----
  gemm_wmma_f32<HID, LAT><<<gemmGrid, 128, 0, stream>>>(z, w2_src, hs);
  gemm_wmma_f32<HID, LAT><<<gemmGrid, 128, 0, stream>>>(z, w2_dst, hd);
  rowdot_kernel<LAT><<<dotGrid, 256, 0, stream>>>(hs, a2_src, alphaS);
  rowdot_kernel<LAT><<<dotGrid, 256, 0, stream>>>(hd, a2_dst, alphaD);

  fill_kernel<<<(N_NODES + 255) / 256, 256, 0, stream>>>(maxb, NEGINF, N_NODES);
  fill_kernel<<<(N_NODES + 255) / 256, 256, 0, stream>>>(denom, 0.f, N_NODES);
  fill_kernel<<<((int)(N_NODES * LAT) + 255) / 256, 256, 0, stream>>>(aggr, 0.f, N_NODES * LAT);
  edge_pass1<<<edgGrid, 256, 0, stream>>>(src, dst, alphaS, alphaD, edgeE, maxb);
  edge_pass2<<<edgGrid, 256, 0, stream>>>(dst, edgeE, maxb, denom);
  edge_pass3<LAT><<<e3Grid, 256, 0, stream>>>(src, dst, edgeE, denom, hs, aggr);

  final_out<<<((int)(N_NODES * LAT) + 255) / 256, 256, 0, stream>>>(aggr, b2, (float*)d_out);
}